// UKFCell_75428215653008
// MI455X (gfx1250) — compile-verified
//
#include <hip/hip_runtime.h>
#include <math.h>

typedef __attribute__((ext_vector_type(2))) float v2f;
typedef __attribute__((ext_vector_type(8))) float v8f;

#define NDIM  32
#define STR   33     // LDS row stride (pad to avoid bank conflicts)
#define NWAVE 4      // waves (= batches) per block
#define NTRI  528    // 32*33/2 linearized tril entries

// One 32x32x32 fp32 GEMM per wave using V_WMMA_F32_16X16X4_F32.
// MODE 0: Dlds = op(A)*op(B) + Cglobal
// MODE 1: Dlds = op(A)*op(B) + Clds
// MODE 2: Dglobal = Clds - op(A)*op(B)
// AT/BT: read A/B transposed from LDS.
template <int MODE, bool AT, bool BT>
__device__ __forceinline__ void wave_gemm32(
    const float* __restrict__ A, const float* __restrict__ Bm,
    const float* __restrict__ Clds, const float* __restrict__ Cglob,
    float* __restrict__ Dlds, float* __restrict__ Dglob,
    int half, int l15)
{
#pragma unroll
  for (int mt = 0; mt < NDIM; mt += 16) {
#pragma unroll
    for (int nt = 0; nt < NDIM; nt += 16) {
      v8f acc;
#pragma unroll
      for (int i = 0; i < 8; ++i) {          // C/D layout: VGPR i -> M = mt+i (+8 for lanes 16-31)
        const int m = mt + i + 8 * half;
        const int n = nt + l15;
        if (MODE == 0)      acc[i] = Cglob[m * NDIM + n];
        else if (MODE == 1) acc[i] = Clds[m * STR + n];
        else                acc[i] = 0.0f;
      }
#pragma unroll
      for (int k0 = 0; k0 < NDIM; k0 += 4) { // A 16x4: lanes 0-15 K={k0,k0+1}, 16-31 K={k0+2,k0+3}
        const int ka = k0 + 2 * half;
        v2f a, b;
        if (!AT) { a.x = A[(mt + l15) * STR + ka];   a.y = A[(mt + l15) * STR + ka + 1]; }
        else     { a.x = A[ka * STR + (mt + l15)];   a.y = A[(ka + 1) * STR + (mt + l15)]; }
        if (!BT) { b.x = Bm[ka * STR + (nt + l15)];  b.y = Bm[(ka + 1) * STR + (nt + l15)]; }
        else     { b.x = Bm[(nt + l15) * STR + ka];  b.y = Bm[(nt + l15) * STR + ka + 1]; }
        acc = __builtin_amdgcn_wmma_f32_16x16x4_f32(false, a, false, b, (short)0, acc,
                                                    false, false);
      }
#pragma unroll
      for (int i = 0; i < 8; ++i) {
        const int m = mt + i + 8 * half;
        const int n = nt + l15;
        if (MODE == 2) Dglob[m * NDIM + n] = Clds[m * STR + n] - acc[i];
        else           Dlds[m * STR + n] = acc[i];
      }
    }
  }
}

__global__ __launch_bounds__(NWAVE * 32) void ukf_fused(
    const float* __restrict__ g_meas, const float* __restrict__ g_state,
    const float* __restrict__ g_scov, const float* __restrict__ g_pn,
    const float* __restrict__ g_mn,
    float* __restrict__ g_y, float* __restrict__ g_ns, float* __restrict__ g_cov,
    int nbatch)
{
  __shared__ float sXC[NWAVE][NDIM * STR];   // x_cov = state_cov + Q
  __shared__ float sYC[NWAVE][NDIM * STR];   // y_cov, then its Cholesky factor C (lower)
  __shared__ float sLB[NWAVE][NDIM * STR];   // Lq -> Lr -> Z

  const int wave = threadIdx.x >> 5;
  const int lane = threadIdx.x & 31;
  int batch = blockIdx.x * NWAVE + wave;
  if (batch >= nbatch) batch = nbatch - 1;   // duplicate work, identical writes: deterministic

  float* XC = sXC[wave];
  float* YC = sYC[wave];
  float* LB = sLB[wave];

  const int half = lane >> 4;
  const int l15  = lane & 15;

  const float* meas = g_meas + (size_t)batch * NDIM;
  const float* st   = g_state + (size_t)batch * NDIM;
  const float* scov = g_scov + (size_t)batch * NDIM * NDIM;
  const float* pn   = g_pn + (size_t)batch * NTRI;
  const float* mn   = g_mn + (size_t)batch * NTRI;

  // y output = state (weights sum to 1, +/- sigma columns cancel exactly)
  const float s0 = st[lane];
  const float d0 = meas[lane] - s0;          // innovation, kept in a register (per-lane component)
  g_y[(size_t)batch * NDIM + lane] = s0;

  // Build Lq (tril, exp on diagonal: LOG_CHOLESKY) -- lane = row
  {
    const int base = (lane * (lane + 1)) >> 1;   // torch.tril_indices row-major start
#pragma unroll
    for (int c = 0; c < NDIM; ++c) {
      float v = 0.0f;
      if (c <= lane) { v = pn[base + c]; if (c == lane) v = expf(v); }
      LB[lane * STR + c] = v;
    }
  }
  __syncthreads();

  // x_cov = Lq*Lq^T + state_cov   (C sourced from global)
  wave_gemm32<0, false, true>(LB, LB, nullptr, scov, XC, nullptr, half, l15);
  __syncthreads();

  // Build Lr
  {
    const int base = (lane * (lane + 1)) >> 1;
#pragma unroll
    for (int c = 0; c < NDIM; ++c) {
      float v = 0.0f;
      if (c <= lane) { v = mn[base + c]; if (c == lane) v = expf(v); }
      LB[lane * STR + c] = v;
    }
  }
  __syncthreads();

  // y_cov = Lr*Lr^T + x_cov
  wave_gemm32<1, false, true>(LB, LB, XC, nullptr, YC, nullptr, half, l15);
  __syncthreads();

  // In-place Cholesky of y_cov (lower triangle). lane = row; column-serial.
  for (int k = 0; k < NDIM; ++k) {
    const float d  = sqrtf(YC[k * STR + k]);
    const float v  = YC[lane * STR + k];
    const float nv = (lane == k) ? d : (v / d);
    if (lane >= k) YC[lane * STR + k] = nv;
    __syncthreads();
    const float lik = nv;
    for (int j = k + 1; j < NDIM; ++j) {
      const float ljk = YC[j * STR + k];        // broadcast read
      if (lane >= j) YC[lane * STR + j] -= lik * ljk;
    }
    __syncthreads();
  }

  // Forward substitution: Z = C^{-1} x_cov.  lane owns RHS column `lane`; z in registers.
  float z[NDIM];
#pragma unroll
  for (int i = 0; i < NDIM; ++i) {
    float s = XC[i * STR + lane];
#pragma unroll
    for (int k = 0; k < i; ++k) s -= YC[i * STR + k] * z[k];   // C[i][k] broadcast
    z[i] = s / YC[i * STR + i];
  }
  // Stage Z into LDS (Z[i][lane] = z[i]) for the WMMA Z^T*Z below
#pragma unroll
  for (int i = 0; i < NDIM; ++i) LB[i * STR + lane] = z[i];

  // Single-RHS solve u = y_cov^{-1} (meas - state) via wave shuffles (no register arrays):
  //   forward:  C t = d    ;  backward:  C^T u = t
  float t = d0;
  for (int k = 0; k < NDIM; ++k) {
    const float tk = __shfl(t, k) / YC[k * STR + k];
    if (lane == k)     t = tk;
    else if (lane > k) t -= YC[lane * STR + k] * tk;
  }
  for (int k = NDIM - 1; k >= 0; --k) {
    const float uk = __shfl(t, k) / YC[k * STR + k];
    if (lane == k)     t = uk;
    else if (lane < k) t -= YC[k * STR + lane] * uk;   // C^T[lane][k] = C[k][lane]
  }
  // new_state = state + x_cov * u   (gain*(meas-state) = x_cov * y_cov^{-1} * d)
  float accns = 0.0f;
#pragma unroll
  for (int j = 0; j < NDIM; ++j) accns += XC[lane * STR + j] * __shfl(t, j);
  g_ns[(size_t)batch * NDIM + lane] = s0 + accns;

  __syncthreads();

  // new_state_cov = x_cov - Z^T * Z   (gain*y_cov*gain^T == Z^T Z), streamed to global
  wave_gemm32<2, true, false>(LB, LB, XC, nullptr, nullptr,
                              g_cov + (size_t)batch * NDIM * NDIM, half, l15);
}

extern "C" void kernel_launch(void* const* d_in, const int* in_sizes, int n_in,
                              void* d_out, int out_size, void* d_ws, size_t ws_size,
                              hipStream_t stream) {
  // setup_inputs order: measurement, state, state_cov, ctrl, process_noise, measurement_noise
  const float* meas  = (const float*)d_in[0];
  const float* state = (const float*)d_in[1];
  const float* scov  = (const float*)d_in[2];
  const float* pn    = (const float*)d_in[4];
  const float* mn    = (const float*)d_in[5];
  const int nbatch = in_sizes[1] / NDIM;

  float* out  = (float*)d_out;                      // (y, new_state, new_state_cov) concat
  float* gy   = out;
  float* gns  = out + (size_t)nbatch * NDIM;
  float* gcov = out + (size_t)2 * nbatch * NDIM;

  const int blocks = (nbatch + NWAVE - 1) / NWAVE;
  ukf_fused<<<blocks, NWAVE * 32, 0, stream>>>(meas, state, scov, pn, mn,
                                               gy, gns, gcov, nbatch);
}